// PaCMAPLoss_41051297415370
// MI455X (gfx1250) — compile-verified
//
#include <hip/hip_runtime.h>

typedef float v2f __attribute__((ext_vector_type(2)));
typedef float v4f __attribute__((ext_vector_type(4)));
typedef float v8f __attribute__((ext_vector_type(8)));

#define PK_GRID  2048
#define PK_BLOCK 256

// ---------------------------------------------------------------------------
// Phase weights, computed from the device-resident iteration scalar exactly as
// the reference does (note: frac uses the static phase constants, == -0.01).
// ---------------------------------------------------------------------------
__device__ __forceinline__ void phase_weights(int it, float& wn, float& wm, float& wf) {
  if (it < 101) {
    const double frac = (0.0 - 1.0) / (101.0 - 1.0);          // -0.01
    wn = 2.0f;
    wm = (float)(1000.0 * (1.0 - frac) + 3.0 * frac);          // 1009.97
    wf = 1.0f;
  } else if (it < 201) {
    wn = 3.0f; wm = 3.0f; wf = 1.0f;
  } else {
    wn = 1.0f; wm = 0.0f; wf = 1.0f;
  }
}

// Per-pair contribution. MODE: 0=near (d/(10+d)), 1=mid (d/(10000+d)), 2=far (1/(1+d))
// with d = |p0-p1|^2 + 1  =>  near: (q+1)/(q+11), mid: (q+1)/(q+10001), far: 1/(q+2)
template <int MODE>
__device__ __forceinline__ float pair_term(v4f p) {
  float dx = p[0] - p[2];
  float dy = p[1] - p[3];
  float q  = fmaf(dx, dx, dy * dy);
  float num = (MODE == 2) ? 1.0f : (q + 1.0f);
  float den = q + ((MODE == 0) ? 11.0f : (MODE == 1) ? 10001.0f : 2.0f);
  return __fdividef(num, den);
}

// Grid-stride over one array with 2x unroll into two independent accumulators
// (these later become the two A-matrix VGPRs of the 16x16x4 f32 WMMA).
template <int MODE>
__device__ __forceinline__ void accum_array(const v4f* __restrict__ p, int n,
                                            long long tid0, long long stride,
                                            float w, float& A0, float& A1) {
  float x0 = 0.0f, x1 = 0.0f;
  long long k = tid0;
  for (; k + stride < (long long)n; k += 2 * stride) {
    x0 += pair_term<MODE>(__builtin_nontemporal_load(&p[k]));
    x1 += pair_term<MODE>(__builtin_nontemporal_load(&p[k + stride]));
  }
  if (k < (long long)n)
    x0 += pair_term<MODE>(__builtin_nontemporal_load(&p[k]));
  A0 = fmaf(w, x0, A0);
  A1 = fmaf(w, x1, A1);
}

// ---------------------------------------------------------------------------
// Main streaming kernel: fused over all three pair arrays; one weighted
// partial per block written to d_ws. Deterministic (fixed reduction tree).
// ---------------------------------------------------------------------------
__global__ void pacmap_partials_kernel(const v4f* __restrict__ p_near, int n_near,
                                       const v4f* __restrict__ p_mid,  int n_mid,
                                       const v4f* __restrict__ p_far,  int n_far,
                                       const int* __restrict__ iter_ptr,
                                       float* __restrict__ partials) {
  float wn, wm, wf;
  phase_weights(*iter_ptr, wn, wm, wf);

  const long long stride = (long long)gridDim.x * blockDim.x;
  const long long tid0   = (long long)blockIdx.x * blockDim.x + threadIdx.x;

  float A0 = 0.0f, A1 = 0.0f;
  accum_array<0>(p_near, n_near, tid0, stride, wn, A0, A1);
  accum_array<1>(p_mid,  n_mid,  tid0, stride, wm, A0, A1);
  accum_array<2>(p_far,  n_far,  tid0, stride, wf, A0, A1);

  // ---- wave32 reduction of the 64 accumulators (A0,A1 per lane) ----
  float t;
#if __has_builtin(__builtin_amdgcn_wmma_f32_16x16x4_f32)
  // A = 16x4 f32 (2 VGPRs/lane), B = all-ones => D[m][n] = sum_k A[m][k].
  // Each lane sums its 8 D VGPRs (covers half the rows), xor-16 combines halves.
  // EXEC is all-1s here: full 256-thread blocks, reconverged after the loops.
  v2f a; a[0] = A0;   a[1] = A1;
  v2f b; b[0] = 1.0f; b[1] = 1.0f;
  v8f c = {};
  c = __builtin_amdgcn_wmma_f32_16x16x4_f32(false, a, false, b, (short)0, c,
                                            false, false);
  t = ((c[0] + c[1]) + (c[2] + c[3])) + ((c[4] + c[5]) + (c[6] + c[7]));
  t += __shfl_xor(t, 16, 32);
#else
  t = A0 + A1;
  for (int off = 16; off > 0; off >>= 1) t += __shfl_down(t, off, 32);
#endif

  __shared__ float s_wave[PK_BLOCK / 32];
  const int lane = threadIdx.x & 31;
  const int wave = threadIdx.x >> 5;
  if (lane == 0) s_wave[wave] = t;
  __syncthreads();
  if (threadIdx.x == 0) {
    float s = 0.0f;
    #pragma unroll
    for (int w = 0; w < PK_BLOCK / 32; ++w) s += s_wave[w];   // fixed order
    partials[blockIdx.x] = s;
  }
}

// ---------------------------------------------------------------------------
// Final deterministic reduction of PK_GRID block partials -> d_out[0].
// ---------------------------------------------------------------------------
__global__ void pacmap_final_kernel(const float* __restrict__ partials, int n,
                                    float* __restrict__ out) {
  float s = 0.0f;
  for (int j = threadIdx.x; j < n; j += blockDim.x) s += partials[j];
  for (int off = 16; off > 0; off >>= 1) s += __shfl_down(s, off, 32);

  __shared__ float s_wave[PK_BLOCK / 32];
  const int lane = threadIdx.x & 31;
  const int wave = threadIdx.x >> 5;
  if (lane == 0) s_wave[wave] = s;
  __syncthreads();
  if (threadIdx.x == 0) {
    float r = 0.0f;
    #pragma unroll
    for (int w = 0; w < PK_BLOCK / 32; ++w) r += s_wave[w];
    out[0] = r;
  }
}

extern "C" void kernel_launch(void* const* d_in, const int* in_sizes, int n_in,
                              void* d_out, int out_size, void* d_ws, size_t ws_size,
                              hipStream_t stream) {
  const v4f* p_near = (const v4f*)d_in[0];   // [5M, 2, 2] f32 -> float4 per pair
  const v4f* p_mid  = (const v4f*)d_in[1];
  const v4f* p_far  = (const v4f*)d_in[2];
  const int* iterp  = (const int*)d_in[3];   // scalar iteration

  const int n_near = in_sizes[0] / 4;
  const int n_mid  = in_sizes[1] / 4;
  const int n_far  = in_sizes[2] / 4;

  float* partials = (float*)d_ws;            // PK_GRID floats (8 KB)

  pacmap_partials_kernel<<<PK_GRID, PK_BLOCK, 0, stream>>>(
      p_near, n_near, p_mid, n_mid, p_far, n_far, iterp, partials);
  pacmap_final_kernel<<<1, PK_BLOCK, 0, stream>>>(partials, PK_GRID,
                                                  (float*)d_out);
}